// PointMaxPool_21715354650019
// MI455X (gfx1250) — compile-verified
//
#include <hip/hip_runtime.h>
#include <float.h>

typedef __attribute__((ext_vector_type(2))) float v2f;
typedef __attribute__((ext_vector_type(8))) float v8f;

#define NPTS  8192
#define MPTS  2048
#define BATCH 8
#define CH    128

// ------------------------------------------------------------------
// Kernel 1: farthest point sampling, one workgroup per batch.
// Positions live in LDS (96KB); running min-distances live in registers
// (8 per thread). Emits sampled_pos directly into d_out's second section.
// ------------------------------------------------------------------
extern "C" __global__ __launch_bounds__(1024)
void fps_kernel(const float* __restrict__ pos,        // (8,3,8192)
                float* __restrict__ sampled_out) {    // (8,3,2048)
  const int b   = blockIdx.x;
  const int tid = threadIdx.x;
  extern __shared__ float smem[];
  float* sx = smem;
  float* sy = smem + NPTS;
  float* sz = smem + 2 * NPTS;
  __shared__ float red_v[32];
  __shared__ int   red_i[32];
  __shared__ int   s_far;

  const float* p = pos + (size_t)b * 3 * NPTS;
  for (int i = tid; i < NPTS; i += 1024) {
    sx[i] = p[i];
    sy[i] = p[NPTS + i];
    sz[i] = p[2 * NPTS + i];
  }
  __syncthreads();

  float dloc[8];
#pragma unroll
  for (int j = 0; j < 8; ++j) dloc[j] = 1e10f;   // matches reference init

  int farthest = 0;
  const int lane = tid & 31;
  const int wave = tid >> 5;

  for (int s = 0; s < MPTS; ++s) {
    if (tid == 0) {
      sampled_out[(size_t)b * 3 * MPTS + 0 * MPTS + s] = sx[farthest];
      sampled_out[(size_t)b * 3 * MPTS + 1 * MPTS + s] = sy[farthest];
      sampled_out[(size_t)b * 3 * MPTS + 2 * MPTS + s] = sz[farthest];
    }
    const float cx = sx[farthest], cy = sy[farthest], cz = sz[farthest];
    float best = -1.0f;
    int   bi   = NPTS;
#pragma unroll
    for (int j = 0; j < 8; ++j) {
      const int i = tid + j * 1024;
      const float dx = sx[i] - cx, dy = sy[i] - cy, dz = sz[i] - cz;
      const float d  = dx * dx + dy * dy + dz * dz;
      const float nd = fminf(dloc[j], d);
      dloc[j] = nd;
      if (nd > best) { best = nd; bi = i; }   // strict > keeps first max
    }
    // wave32 butterfly argmax (tie -> smaller index, matching jnp.argmax)
#pragma unroll
    for (int off = 16; off > 0; off >>= 1) {
      const float ov = __shfl_xor(best, off, 32);
      const int   oi = __shfl_xor(bi,   off, 32);
      if (ov > best || (ov == best && oi < bi)) { best = ov; bi = oi; }
    }
    if (lane == 0) { red_v[wave] = best; red_i[wave] = bi; }
    __syncthreads();
    if (wave == 0) {
      float v  = red_v[lane];
      int   ii = red_i[lane];
#pragma unroll
      for (int off = 16; off > 0; off >>= 1) {
        const float ov = __shfl_xor(v,  off, 32);
        const int   oi = __shfl_xor(ii, off, 32);
        if (ov > v || (ov == v && oi < ii)) { v = ov; ii = oi; }
      }
      if (lane == 0) s_far = ii;
    }
    __syncthreads();
    farthest = s_far;
  }
}

// ------------------------------------------------------------------
// Kernel 2: p2[b,n] = |pos|^2 (ranking term that varies per column)
// ------------------------------------------------------------------
extern "C" __global__ void p2_kernel(const float* __restrict__ pos,
                                     float* __restrict__ p2) {
  const int t = blockIdx.x * blockDim.x + threadIdx.x;
  if (t >= BATCH * NPTS) return;
  const int b = t / NPTS, i = t - b * NPTS;
  const float* p = pos + (size_t)b * 3 * NPTS;
  const float x = p[i], y = p[NPTS + i], z = p[2 * NPTS + i];
  p2[t] = x * x + y * y + z * z;
}

// ------------------------------------------------------------------
// Kernel 3: fused WMMA distance GEMM + top-4 + feature max-pool.
// One wave per 16-row m-tile; n swept in 16-col V_WMMA_F32_16X16X4_F32
// steps (K=3 padded to 4). Score = p2[n] - 2*cross (s2 row-constant,
// irrelevant for per-row ranking).
// ------------------------------------------------------------------
__device__ __forceinline__ void ins4(float d, int g,
                                     float& t0, float& t1, float& t2, float& t3,
                                     int& i0, int& i1, int& i2, int& i3) {
  if (d < t3) {
    if (d < t2) {
      t3 = t2; i3 = i2;
      if (d < t1) {
        t2 = t1; i2 = i1;
        if (d < t0) { t1 = t0; i1 = i0; t0 = d; i0 = g; }
        else        { t1 = d;  i1 = g; }
      } else { t2 = d; i2 = g; }
    } else { t3 = d; i3 = g; }
  }
}

extern "C" __global__ __launch_bounds__(256)
void knn_pool_kernel(const float* __restrict__ feats,    // (8,128,8192)
                     const float* __restrict__ sampled,  // (8,3,2048)
                     const float* __restrict__ pos,      // (8,3,8192)
                     const float* __restrict__ p2g,      // (8,8192)
                     float* __restrict__ pooled) {       // (8,128,2048)
  const int b    = blockIdx.y;
  const int wave = threadIdx.x >> 5;
  const int lane = threadIdx.x & 31;
  const int m0   = (blockIdx.x * 8 + wave) * 16;         // 16-row tile base

  __shared__ __align__(16) float tile[8][256];  // per-wave 16x16 score tile
  __shared__ int nn[8][16][4];                  // per-wave top-4 indices per row

  const float* S  = sampled + (size_t)b * 3 * MPTS;
  const float* P  = pos     + (size_t)b * 3 * NPTS;
  const float* PQ = p2g     + (size_t)b * NPTS;

  const int half = lane >> 4;      // K split: lanes 0-15 hold K0,K1; 16-31 hold K2,pad
  const int lrow = lane & 15;

  // A matrix: 16x4 fp32 (sampled xyz, K padded with 0). Branchless K-split:
  // lane half h loads plane 2h (x or z) and plane 1 (y), pad y with 0 for h==1.
  const float a0   = S[half * 2 * MPTS + m0 + lrow];
  const float a1_r = S[MPTS + m0 + lrow];
  v2f a;
  a.x = a0;
  a.y = half ? 0.0f : a1_r;

  float t0 = FLT_MAX, t1 = FLT_MAX, t2 = FLT_MAX, t3 = FLT_MAX;
  int   i0 = 0, i1 = 0, i2 = 0, i3 = 0;

  // Incremented pointers: per-chunk addressing is pure immediate-offset.
  const float* pB0 = P + half * 2 * NPTS + lrow;   // x-plane (h=0) or z-plane (h=1)
  const float* pB1 = P + NPTS + lrow;              // y-plane
  const float* pPQ = PQ + lrow;
  float* const myTileW = &tile[wave][0];                       // D-layout writes
  const float4* const myTileR =
      (const float4*)&tile[wave][lrow * 16 + half * 8];        // row-scan reads

  for (int n0 = 0; n0 < NPTS; n0 += 16, pB0 += 16, pB1 += 16, pPQ += 16) {
    // B matrix: 4x16 fp32, branchless K-split (no exec divergence)
    const float b0   = pB0[0];
    const float b1_r = pB1[0];
    v2f bm;
    bm.x = b0;
    bm.y = half ? 0.0f : b1_r;

    v8f c = {0.f, 0.f, 0.f, 0.f, 0.f, 0.f, 0.f, 0.f};
    c = __builtin_amdgcn_wmma_f32_16x16x4_f32(false, a, false, bm,
                                              (short)0, c, false, false);

    const float pq = pPQ[0];
#pragma unroll
    for (int v = 0; v < 8; ++v) {
      const int rowM = v + 8 * half;                 // D layout: M = v + 8*half
      myTileW[rowM * 16 + lrow] = pq - 2.0f * c[v];
    }
    // per-wave LDS ops are in-order; force completion + block reordering
    asm volatile("s_wait_dscnt 0" ::: "memory");

    // Each lane scans half a row: two b128 LDS loads + min-reduce fast reject.
    const float4 ra = myTileR[0];
    const float4 rb = myTileR[1];
    const float mn = fminf(fminf(fminf(ra.x, ra.y), fminf(ra.z, ra.w)),
                           fminf(fminf(rb.x, rb.y), fminf(rb.z, rb.w)));
    if (mn < t3) {                                   // rare path after warm-up
      const int base = n0 + half * 8;
      ins4(ra.x, base + 0, t0, t1, t2, t3, i0, i1, i2, i3);
      ins4(ra.y, base + 1, t0, t1, t2, t3, i0, i1, i2, i3);
      ins4(ra.z, base + 2, t0, t1, t2, t3, i0, i1, i2, i3);
      ins4(ra.w, base + 3, t0, t1, t2, t3, i0, i1, i2, i3);
      ins4(rb.x, base + 4, t0, t1, t2, t3, i0, i1, i2, i3);
      ins4(rb.y, base + 5, t0, t1, t2, t3, i0, i1, i2, i3);
      ins4(rb.z, base + 6, t0, t1, t2, t3, i0, i1, i2, i3);
      ins4(rb.w, base + 7, t0, t1, t2, t3, i0, i1, i2, i3);
    }
    asm volatile("" ::: "memory");
  }

  // merge the two half-row candidate lists (lane ^ 16), both sorted ascending
  const float p0 = __shfl_xor(t0, 16, 32), p1 = __shfl_xor(t1, 16, 32);
  const float p2v = __shfl_xor(t2, 16, 32), p3 = __shfl_xor(t3, 16, 32);
  const int   q0 = __shfl_xor(i0, 16, 32), q1 = __shfl_xor(i1, 16, 32);
  const int   q2 = __shfl_xor(i2, 16, 32), q3 = __shfl_xor(i3, 16, 32);
  ins4(p0, q0, t0, t1, t2, t3, i0, i1, i2, i3);
  ins4(p1, q1, t0, t1, t2, t3, i0, i1, i2, i3);
  ins4(p2v, q2, t0, t1, t2, t3, i0, i1, i2, i3);
  ins4(p3, q3, t0, t1, t2, t3, i0, i1, i2, i3);

  if (half == 0) {
    nn[wave][lrow][0] = i0; nn[wave][lrow][1] = i1;
    nn[wave][lrow][2] = i2; nn[wave][lrow][3] = i3;
  }
  asm volatile("s_wait_dscnt 0" ::: "memory");

  // fused max-pool over the 4 neighbors; lane owns row lrow, half the channels
  const int r  = lrow;
  const int j0 = nn[wave][r][0], j1 = nn[wave][r][1];
  const int j2 = nn[wave][r][2], j3 = nn[wave][r][3];
  const float* Fb = feats + (size_t)b * CH * NPTS;
  float* Ob = pooled + (size_t)b * CH * MPTS;
  for (int ch = half; ch < CH; ch += 2) {
    const float* F = Fb + (size_t)ch * NPTS;
    const float m = fmaxf(fmaxf(F[j0], F[j1]), fmaxf(F[j2], F[j3]));
    Ob[(size_t)ch * MPTS + (m0 + r)] = m;
  }
}

// ------------------------------------------------------------------
extern "C" void kernel_launch(void* const* d_in, const int* in_sizes, int n_in,
                              void* d_out, int out_size, void* d_ws, size_t ws_size,
                              hipStream_t stream) {
  const float* feats = (const float*)d_in[0];   // (8,128,8192)
  const float* pos   = (const float*)d_in[1];   // (8,3,8192)

  float* pooled  = (float*)d_out;                              // 8*128*2048
  float* sampled = (float*)d_out + (size_t)BATCH * CH * MPTS;  // 8*3*2048
  float* p2      = (float*)d_ws;                               // 8*8192 (256KB)

  fps_kernel<<<BATCH, 1024, 3 * NPTS * sizeof(float), stream>>>(pos, sampled);
  p2_kernel<<<(BATCH * NPTS + 255) / 256, 256, 0, stream>>>(pos, p2);
  knn_pool_kernel<<<dim3(16, BATCH), 256, 0, stream>>>(feats, sampled, pos, p2, pooled);
}